// CSMamba_63874753626316
// MI455X (gfx1250) — compile-verified
//
#include <hip/hip_runtime.h>

typedef __attribute__((ext_vector_type(2))) float v2f;
typedef __attribute__((ext_vector_type(8))) float v8f;

#define EPSF 1e-5f
#define BATCH_C 256
#define SPA_LEN_C 100
#define SPE_LEN_C 27

__device__ __forceinline__ float sigmoid_(float x) { return 1.0f / (1.0f + __expf(-x)); }
__device__ __forceinline__ float softplus_(float x) { return (x > 20.0f) ? x : log1pf(__expf(x)); }

// ---------------------------------------------------------------------------
// WMMA helpers: D += A(16xK) * W(16,K)^T using V_WMMA_F32_16X16X4_F32.
// A-frag: lanes 0-15 row m, cols {k,k+1}; lanes 16-31 row m, cols {k+2,k+3}.
// B-frag: B = W^T, so lane n reads W[n][k..k+1] (lanes 16-31: k+2..k+3).
// C/D: VGPR j -> row m0 + j + 8*half, col n0 + (lane&15).
// ---------------------------------------------------------------------------
template <int K>
__device__ __forceinline__ v8f wmma_kloop(const float* arow, const float* wrow, v8f acc) {
#pragma unroll
  for (int k = 0; k < K; k += 4) {
    v2f a = *(const v2f*)(arow + k);
    v2f b = *(const v2f*)(wrow + k);
    acc = __builtin_amdgcn_wmma_f32_16x16x4_f32(false, a, false, b, (short)0, acc, false, false);
  }
  return acc;
}

// A in global memory, rows clamped to [0, Trows-1]
template <int K>
__device__ __forceinline__ v8f wmma_tile_g(const float* __restrict__ A, int lda, int m0, int Trows,
                                           const float* __restrict__ W, int ldw, v8f acc) {
  const int lane = threadIdx.x & 31;
  const int half = lane >> 4;
  const int r = lane & 15;
  int row = m0 + r; row = (row < Trows) ? row : (Trows - 1);
  return wmma_kloop<K>(A + (size_t)row * lda + 2 * half, W + (size_t)r * ldw + 2 * half, acc);
}

// A in LDS (generic pointer), lda = padded row stride (even)
template <int K>
__device__ __forceinline__ v8f wmma_tile_s(const float* A, int lda,
                                           const float* __restrict__ W, int ldw, v8f acc) {
  const int lane = threadIdx.x & 31;
  const int half = lane >> 4;
  const int r = lane & 15;
  return wmma_kloop<K>(A + r * lda + 2 * half, W + (size_t)r * ldw + 2 * half, acc);
}

__device__ __forceinline__ void store_tile(float* __restrict__ out, int ldo, int m0, int n0,
                                           v8f acc, const float* __restrict__ bias, int Trows) {
  const int lane = threadIdx.x & 31;
  const int half = lane >> 4;
  const int r = lane & 15;
  float bv = bias ? bias[n0 + r] : 0.0f;
#pragma unroll
  for (int j = 0; j < 8; ++j) {
    int m = m0 + j + 8 * half;
    if (m < Trows) out[(size_t)m * ldo + n0 + r] = acc[j] + bv;
  }
}

// ---------------------------------------------------------------------------
// Wcomb = dt_proj.w (128x4) @ x_proj_w[0:4,:] (4x128)  -> (128x128)
// ---------------------------------------------------------------------------
__global__ void k_wcomb(const float* __restrict__ dtw, const float* __restrict__ xpw,
                        float* __restrict__ wc) {
  int idx = blockIdx.x * blockDim.x + threadIdx.x;  // 16384 threads
  int i = idx >> 7, j = idx & 127;
  float s = 0.0f;
#pragma unroll
  for (int r = 0; r < 4; ++r) s += dtw[i * 4 + r] * xpw[r * 128 + j];
  wc[idx] = s;
}

// ---------------------------------------------------------------------------
// RMSNorm 16 tokens into LDS, then in_proj GEMM (64 -> 256) via WMMA.
// ---------------------------------------------------------------------------
__global__ void k_rms_inproj(const float* __restrict__ x, const float* __restrict__ rmsw,
                             const float* __restrict__ Win, const float* __restrict__ bin,
                             float* __restrict__ xn, float* __restrict__ xz, int T) {
  __shared__ float xs[16][66];
  const int tid = threadIdx.x;
  const int m0 = blockIdx.x * 16;
  {
    int tt = tid >> 4;         // token within tile (half-wave per token)
    int l16 = tid & 15;
    int t = m0 + tt; t = (t < T) ? t : (T - 1);
    float v[4]; float ss = 0.0f;
#pragma unroll
    for (int i = 0; i < 4; ++i) { v[i] = x[(size_t)t * 64 + l16 * 4 + i]; ss += v[i] * v[i]; }
#pragma unroll
    for (int o = 8; o >= 1; o >>= 1) ss += __shfl_xor(ss, o, 16);
    float sc = rsqrtf(ss * (1.0f / 64.0f) + EPSF);
#pragma unroll
    for (int i = 0; i < 4; ++i) {
      int c = l16 * 4 + i;
      float nv = v[i] * sc * rmsw[c];
      xs[tt][c] = nv;
      if (m0 + tt < T) xn[(size_t)(m0 + tt) * 64 + c] = nv;
    }
  }
  __syncthreads();
  const int w = tid >> 5;  // wave id 0..7
#pragma unroll
  for (int s = 0; s < 2; ++s) {
    int nt = w + 8 * s;  // 16 N-tiles -> out = 256
    v8f acc = {};
    acc = wmma_tile_s<64>(&xs[0][0], 66, Win + (size_t)nt * 16 * 64, 64, acc);
    store_tile(xz, 256, m0, nt * 16, acc, bin, T);
  }
}

// ---------------------------------------------------------------------------
// Causal depthwise conv (k=4) + SiLU on first 128 channels of xz.
// ---------------------------------------------------------------------------
__global__ void k_conv_silu(const float* __restrict__ xz, const float* __restrict__ cw,
                            const float* __restrict__ cb, float* __restrict__ xm, int T, int L) {
  int idx = blockIdx.x * blockDim.x + threadIdx.x;
  if (idx >= T * 128) return;
  int d = idx & 127;
  int t = idx >> 7;
  int l = t % L;
  float acc = cb[d];
#pragma unroll
  for (int k = 0; k < 4; ++k) {
    int ls = l - 3 + k;
    if (ls >= 0) acc += cw[d * 4 + k] * xz[(size_t)(t - 3 + k) * 256 + d];
  }
  xm[(size_t)t * 128 + d] = acc * sigmoid_(acc);
}

// ---------------------------------------------------------------------------
// dt = softplus(xm @ Wcomb^T + dtb) (8 tiles); BC = xm @ x_proj_w[4:36]^T (2 tiles).
// ---------------------------------------------------------------------------
__global__ void k_xproj(const float* __restrict__ xm, const float* __restrict__ xpw,
                        const float* __restrict__ wc, const float* __restrict__ dtb,
                        float* __restrict__ bcout, float* __restrict__ dt, int T) {
  const int tid = threadIdx.x;
  const int m0 = blockIdx.x * 16;
  const int w = tid >> 5;
  {
    v8f acc = {};
    acc = wmma_tile_g<128>(xm, 128, m0, T, wc + (size_t)w * 16 * 128, 128, acc);
    const int lane = tid & 31, half = lane >> 4, r = lane & 15;
    float bv = dtb[w * 16 + r];
#pragma unroll
    for (int j = 0; j < 8; ++j) {
      int m = m0 + j + 8 * half;
      if (m < T) dt[(size_t)m * 128 + w * 16 + r] = softplus_(acc[j] + bv);
    }
  }
  if (w < 2) {  // wave-uniform: EXEC all-ones inside
    v8f acc = {};
    acc = wmma_tile_g<128>(xm, 128, m0, T, xpw + (size_t)(4 + w * 16) * 128, 128, acc);
    store_tile(bcout, 32, m0, w * 16, acc, nullptr, T);
  }
}

// ---------------------------------------------------------------------------
// Selective scan: one block per batch. thread -> (d = tid/2, 8 states).
// ---------------------------------------------------------------------------
__global__ void k_scan(const float* __restrict__ dt, const float* __restrict__ xm,
                       const float* __restrict__ bc, const float* __restrict__ xz,
                       const float* __restrict__ alog, const float* __restrict__ Dp,
                       float* __restrict__ y, int L) {
  const int b = blockIdx.x;
  const int tid = threadIdx.x;
  const int d = tid >> 1;
  const int no = (tid & 1) * 8;
  float A[8], h[8];
#pragma unroll
  for (int j = 0; j < 8; ++j) { A[j] = -__expf(alog[d * 16 + no + j]); h[j] = 0.0f; }
  const float Dv = Dp[d];
  for (int l = 0; l < L; ++l) {
    size_t t = (size_t)b * L + l;
    float dtv = dt[t * 128 + d];
    float xmv = xm[t * 128 + d];
    float ys = 0.0f;
#pragma unroll
    for (int j = 0; j < 8; ++j) {
      float Bv = bc[t * 32 + no + j];
      float Cv = bc[t * 32 + 16 + no + j];
      h[j] = __expf(dtv * A[j]) * h[j] + dtv * Bv * xmv;
      ys += h[j] * Cv;
    }
    ys += __shfl_xor(ys, 1, 32);  // combine the two 8-state halves of channel d
    float z = xz[t * 256 + 128 + d];
    y[t * 128 + d] = (ys + xmv * Dv) * z * sigmoid_(z);
  }
}

// ---------------------------------------------------------------------------
// blk = xn + (y @ Wout^T + bout) @ Wl^T + bl   (chained WMMA, LDS intermediate)
// ---------------------------------------------------------------------------
__global__ void k_outproj(const float* __restrict__ y, const float* __restrict__ xn,
                          const float* __restrict__ Wout, const float* __restrict__ bout,
                          const float* __restrict__ Wl, const float* __restrict__ bl,
                          float* __restrict__ blkout, int T) {
  __shared__ float us[16][66];
  const int tid = threadIdx.x;  // 128 threads, 4 waves
  const int m0 = blockIdx.x * 16;
  const int w = tid >> 5;
  {
    v8f acc = {};
    acc = wmma_tile_g<128>(y, 128, m0, T, Wout + (size_t)w * 16 * 128, 128, acc);
    const int lane = tid & 31, half = lane >> 4, r = lane & 15;
    float bv = bout[w * 16 + r];
#pragma unroll
    for (int j = 0; j < 8; ++j) us[j + 8 * half][w * 16 + r] = acc[j] + bv;
  }
  __syncthreads();
  {
    v8f acc = {};
    acc = wmma_tile_s<64>(&us[0][0], 66, Wl + (size_t)w * 16 * 64, 64, acc);
    const int lane = tid & 31, half = lane >> 4, r = lane & 15;
    float bv = bl[w * 16 + r];
#pragma unroll
    for (int j = 0; j < 8; ++j) {
      int m = m0 + j + 8 * half;
      if (m < T) blkout[(size_t)m * 64 + w * 16 + r] =
                     xn[(size_t)m * 64 + w * 16 + r] + acc[j] + bv;
    }
  }
}

// ---------------------------------------------------------------------------
// Star gate: ss = relu((s0@W1^T+b1) * (s0@W2^T+b2)), s0 = spa_center + mean(spe)
// ---------------------------------------------------------------------------
__global__ void k_ss(const float* __restrict__ spa_blk, const float* __restrict__ spe_blk,
                     const float* __restrict__ w1, const float* __restrict__ b1,
                     const float* __restrict__ w2, const float* __restrict__ b2,
                     float* __restrict__ ss) {
  __shared__ float s0[16][66];
  __shared__ float t1[16][66];
  __shared__ float t2[16][66];
  const int tid = threadIdx.x;
  const int b0 = blockIdx.x * 16;
#pragma unroll
  for (int i = 0; i < 4; ++i) {
    int e = tid + i * 256;
    int bb = e >> 6, c = e & 63;
    int b = b0 + bb;
    float acc = spa_blk[((size_t)b * SPA_LEN_C + (SPA_LEN_C / 2 + 1)) * 64 + c];
    float mn = 0.0f;
    for (int l = 0; l < SPE_LEN_C; ++l) mn += spe_blk[((size_t)b * SPE_LEN_C + l) * 64 + c];
    s0[bb][c] = acc + mn * (1.0f / SPE_LEN_C);
  }
  __syncthreads();
  const int w = tid >> 5;
  {
    const float* W = (w < 4) ? w1 : w2;
    const float* B = (w < 4) ? b1 : b2;
    int nt = w & 3;
    v8f acc = {};
    acc = wmma_tile_s<64>(&s0[0][0], 66, W + (size_t)nt * 16 * 64, 64, acc);
    const int lane = tid & 31, half = lane >> 4, r = lane & 15;
    float bv = B[nt * 16 + r];
    float(*dst)[66] = (w < 4) ? t1 : t2;
#pragma unroll
    for (int j = 0; j < 8; ++j) dst[j + 8 * half][nt * 16 + r] = acc[j] + bv;
  }
  __syncthreads();
#pragma unroll
  for (int i = 0; i < 4; ++i) {
    int e = tid + i * 256;
    int bb = e >> 6, c = e & 63;
    float v = t1[bb][c] * t2[bb][c];
    ss[(size_t)(b0 + bb) * 64 + c] = (v > 0.0f) ? v : 0.0f;
  }
}

// ---------------------------------------------------------------------------
// out = LayerNorm((blk*ss) @ Wc^T + bc) + silu(x_orig @ Wl^T + bl)
// ---------------------------------------------------------------------------
__global__ void k_final(const float* __restrict__ blk, const float* __restrict__ xorig,
                        const float* __restrict__ ss, const float* __restrict__ Wc,
                        const float* __restrict__ bcv, const float* __restrict__ lnw,
                        const float* __restrict__ lnb, const float* __restrict__ Wl,
                        const float* __restrict__ bl, float* __restrict__ out, int T, int L) {
  __shared__ float as[16][66];
  __shared__ float gs[16][66];
  __shared__ float hs[16][66];
  const int tid = threadIdx.x;
  const int m0 = blockIdx.x * 16;
#pragma unroll
  for (int i = 0; i < 4; ++i) {
    int e = tid + i * 256;
    int m = e >> 6, c = e & 63;
    int t = m0 + m; t = (t < T) ? t : (T - 1);
    int b = t / L;
    as[m][c] = blk[(size_t)t * 64 + c] * ss[(size_t)b * 64 + c];
  }
  __syncthreads();
  const int w = tid >> 5;
  if (w < 4) {  // conv-path GEMM from LDS
    v8f acc = {};
    acc = wmma_tile_s<64>(&as[0][0], 66, Wc + (size_t)w * 16 * 64, 64, acc);
    const int lane = tid & 31, half = lane >> 4, r = lane & 15;
    float bv = bcv[w * 16 + r];
#pragma unroll
    for (int j = 0; j < 8; ++j) gs[j + 8 * half][w * 16 + r] = acc[j] + bv;
  } else {  // silu-path GEMM straight from global input
    int nt = w - 4;
    v8f acc = {};
    acc = wmma_tile_g<64>(xorig, 64, m0, T, Wl + (size_t)nt * 16 * 64, 64, acc);
    const int lane = tid & 31, half = lane >> 4, r = lane & 15;
    float bv = bl[nt * 16 + r];
#pragma unroll
    for (int j = 0; j < 8; ++j) {
      float v = acc[j] + bv;
      hs[j + 8 * half][nt * 16 + r] = v * sigmoid_(v);
    }
  }
  __syncthreads();
  {  // per-row LayerNorm (16 lanes per row, 4 cols each) + combine
    int m = tid >> 4, l16 = tid & 15;
    float v[4]; float s = 0.0f;
#pragma unroll
    for (int i = 0; i < 4; ++i) { v[i] = gs[m][l16 * 4 + i]; s += v[i]; }
#pragma unroll
    for (int o = 8; o >= 1; o >>= 1) s += __shfl_xor(s, o, 16);
    float mu = s * (1.0f / 64.0f);
    float q = 0.0f;
#pragma unroll
    for (int i = 0; i < 4; ++i) { float d0 = v[i] - mu; q += d0 * d0; }
#pragma unroll
    for (int o = 8; o >= 1; o >>= 1) q += __shfl_xor(q, o, 16);
    float inv = rsqrtf(q * (1.0f / 64.0f) + EPSF);
    int t = m0 + m;
    if (t < T) {
#pragma unroll
      for (int i = 0; i < 4; ++i) {
        int c = l16 * 4 + i;
        out[(size_t)t * 64 + c] = (v[i] - mu) * inv * lnw[c] + lnb[c] + hs[m][c];
      }
    }
  }
}

// ---------------------------------------------------------------------------
// Host launcher. Input leaves assumed flattened in setup_inputs() insertion
// order: spa, spe, [spa_mamba 14 leaves], [spe_mamba 14 leaves], fusion 12, top 4.
// ---------------------------------------------------------------------------
extern "C" void kernel_launch(void* const* d_in, const int* in_sizes, int n_in,
                              void* d_out, int out_size, void* d_ws, size_t ws_size,
                              hipStream_t stream) {
  (void)in_sizes; (void)n_in; (void)out_size; (void)ws_size;
  const float* spa_x = (const float*)d_in[0];
  const float* spe_x = (const float*)d_in[1];
  const float* P1[14]; const float* P2[14];
  for (int i = 0; i < 14; ++i) {
    P1[i] = (const float*)d_in[2 + i];    // rms_w, in_w, in_b, conv_w, conv_b, xp_w,
    P2[i] = (const float*)d_in[16 + i];   // dt_w, dt_b, A_log, D, out_w, out_b, l1_w, l1_b
  }
  auto F = [&](int i) { return (const float*)d_in[i]; };

  const int T1 = BATCH_C * SPA_LEN_C;  // 25600
  const int T2 = BATCH_C * SPE_LEN_C;  // 6912

  float* ws = (float*)d_ws;
  size_t off = 0;
  auto alloc = [&](size_t n) { float* p = ws + off; off += (n + 1) & ~(size_t)1; return p; };
  // shared between spa/spe passes (stream-serialized), sized for the larger (spa)
  float* xn    = alloc((size_t)T1 * 64);
  float* xz    = alloc((size_t)T1 * 256);
  float* xm    = alloc((size_t)T1 * 128);
  float* dt    = alloc((size_t)T1 * 128);
  float* bc    = alloc((size_t)T1 * 32);
  float* yb    = alloc((size_t)T1 * 128);
  float* blk1  = alloc((size_t)T1 * 64);
  float* blk2  = alloc((size_t)T2 * 64);
  float* wcomb = alloc(128 * 128);
  float* ssb   = alloc((size_t)BATCH_C * 64);

  auto run_mamba = [&](const float* x, int T, int L, const float* const* P, float* blk) {
    int tiles = (T + 15) / 16;
    k_wcomb<<<64, 256, 0, stream>>>(P[6], P[5], wcomb);
    k_rms_inproj<<<tiles, 256, 0, stream>>>(x, P[0], P[1], P[2], xn, xz, T);
    k_conv_silu<<<(T * 128 + 255) / 256, 256, 0, stream>>>(xz, P[3], P[4], xm, T, L);
    k_xproj<<<tiles, 256, 0, stream>>>(xm, P[5], wcomb, P[7], bc, dt, T);
    k_scan<<<BATCH_C, 256, 0, stream>>>(dt, xm, bc, xz, P[8], P[9], yb, L);
    k_outproj<<<tiles, 128, 0, stream>>>(yb, xn, P[10], P[11], P[12], P[13], blk, T);
  };
  run_mamba(spa_x, T1, SPA_LEN_C, P1, blk1);
  run_mamba(spe_x, T2, SPE_LEN_C, P2, blk2);

  k_ss<<<BATCH_C / 16, 256, 0, stream>>>(blk1, blk2, F(30), F(31), F(32), F(33), ssb);

  float* out_spa = (float*)d_out;
  float* out_spe = out_spa + (size_t)T1 * 64;
  k_final<<<(T1 + 15) / 16, 256, 0, stream>>>(blk1, spa_x, ssb, F(34), F(35), F(38), F(39),
                                              F(42), F(43), out_spa, T1, SPA_LEN_C);
  k_final<<<(T2 + 15) / 16, 256, 0, stream>>>(blk2, spe_x, ssb, F(36), F(37), F(40), F(41),
                                              F(44), F(45), out_spe, T2, SPE_LEN_C);
}